// DenseGrid_55645596287135
// MI455X (gfx1250) — compile-verified
//
#include <hip/hip_runtime.h>

// Fused NGLOD dense-grid trilerp, 5 LODs (16,32,64,128,256), F=3, ReLU per LOD.
// Memory-bound gather kernel: LOD0 table staged in LDS via CDNA5 async
// global->LDS DMA; all other gathers are 12-byte (b96) vector loads that hit
// the 192MB L2 for LODs 1-3 and mostly-resident LOD4.

namespace {

struct F3 { float x, y, z; };

constexpr int BLOCK = 256;
constexpr int R0 = 16;
constexpr int LDS_FLOATS   = R0 * R0 * R0 * 3;        // 12288 floats = 48 KiB
constexpr int LDS_CHUNKS16 = (LDS_FLOATS * 4) / 16;   // 3072 x 16-byte chunks

// Global-memory corner fetch: one 12-byte load (global_load_b96).
struct GlobalGet {
  const F3* __restrict__ g;
  __device__ __forceinline__ void operator()(int id, float& v0, float& v1, float& v2) const {
    F3 v = g[id];
    v0 = v.x; v1 = v.y; v2 = v.z;
  }
};

template <int RES, class GetF>
__device__ __forceinline__ void lod_accumulate(GetF get, float px, float py, float pz,
                                               float& a0, float& a1, float& a2) {
  const float s  = 0.5f * (float)(RES - 1);
  const float hi = (float)(RES - 1) - 1e-05f;
  // Raw (unclipped) scaled coords are used for the weights, matching reference.
  float x = (px + 1.0f) * s;
  float y = (py + 1.0f) * s;
  float z = (pz + 1.0f) * s;
  int x1 = (int)floorf(fminf(fmaxf(x, 0.0f), hi));
  int y1 = (int)floorf(fminf(fmaxf(y, 0.0f), hi));
  int z1 = (int)floorf(fminf(fmaxf(z, 0.0f), hi));
  int x2 = min(x1 + 1, RES - 1);
  int y2 = min(y1 + 1, RES - 1);
  int z2 = min(z1 + 1, RES - 1);
  float wx0 = (float)x2 - x, wx1 = x - (float)x1;
  float wy0 = (float)y2 - y, wy1 = y - (float)y1;
  float wz0 = (float)z2 - z, wz1 = z - (float)z1;
  const int r2 = RES * RES;
  const int b00 = z1 * r2 + y1 * RES;
  const int b01 = z1 * r2 + y2 * RES;
  const int b10 = z2 * r2 + y1 * RES;
  const int b11 = z2 * r2 + y2 * RES;

  float s0 = 0.0f, s1 = 0.0f, s2 = 0.0f;
  float v0, v1, v2;
  get(b00 + x1, v0, v1, v2); { float w = wx0 * wy0 * wz0; s0 += w * v0; s1 += w * v1; s2 += w * v2; }
  get(b00 + x2, v0, v1, v2); { float w = wx1 * wy0 * wz0; s0 += w * v0; s1 += w * v1; s2 += w * v2; }
  get(b01 + x1, v0, v1, v2); { float w = wx0 * wy1 * wz0; s0 += w * v0; s1 += w * v1; s2 += w * v2; }
  get(b01 + x2, v0, v1, v2); { float w = wx1 * wy1 * wz0; s0 += w * v0; s1 += w * v1; s2 += w * v2; }
  get(b10 + x1, v0, v1, v2); { float w = wx0 * wy0 * wz1; s0 += w * v0; s1 += w * v1; s2 += w * v2; }
  get(b10 + x2, v0, v1, v2); { float w = wx1 * wy0 * wz1; s0 += w * v0; s1 += w * v1; s2 += w * v2; }
  get(b11 + x1, v0, v1, v2); { float w = wx0 * wy1 * wz1; s0 += w * v0; s1 += w * v1; s2 += w * v2; }
  get(b11 + x2, v0, v1, v2); { float w = wx1 * wy1 * wz1; s0 += w * v0; s1 += w * v1; s2 += w * v2; }

  a0 += fmaxf(s0, 0.0f);
  a1 += fmaxf(s1, 0.0f);
  a2 += fmaxf(s2, 0.0f);
}

__global__ void __launch_bounds__(BLOCK) dense_grid_fused(
    const float* __restrict__ pts,
    const float* __restrict__ cb0, const float* __restrict__ cb1,
    const float* __restrict__ cb2, const float* __restrict__ cb3,
    const float* __restrict__ cb4,
    float* __restrict__ out, int N) {
  __shared__ float g0[LDS_FLOATS];  // 48 KiB: entire LOD0 table

  // --- Stage LOD0 into LDS with async global->LDS DMA (ASYNCcnt path). ---
  {
    const unsigned long long gbase = (unsigned long long)(const void*)cb0;
    const unsigned int       lbase = (unsigned int)(unsigned long long)(void*)g0;
    #pragma unroll 4
    for (int c = threadIdx.x; c < LDS_CHUNKS16; c += BLOCK) {
      unsigned long long ga = gbase + (unsigned long long)c * 16ull;
      unsigned int       la = lbase + (unsigned int)c * 16u;
      // vdst = LDS byte address, vaddr = 64-bit global address
      asm volatile("global_load_async_to_lds_b128 %0, %1, off"
                   :: "v"(la), "v"(ga) : "memory");
    }
    asm volatile("s_wait_asynccnt 0" ::: "memory");
  }
  __syncthreads();

  const int n = blockIdx.x * BLOCK + threadIdx.x;
  if (n >= N) return;

  F3 p = reinterpret_cast<const F3*>(pts)[n];  // global_load_b96

  float a0 = 0.0f, a1 = 0.0f, a2 = 0.0f;

  // LOD0 from LDS (ds_load), LODs 1..4 from global via 12B vector gathers.
  auto ldsGet = [&](int id, float& v0, float& v1, float& v2) {
    int b = id * 3;
    v0 = g0[b + 0]; v1 = g0[b + 1]; v2 = g0[b + 2];
  };
  lod_accumulate<16 >(ldsGet, p.x, p.y, p.z, a0, a1, a2);
  lod_accumulate<32 >(GlobalGet{reinterpret_cast<const F3*>(cb1)}, p.x, p.y, p.z, a0, a1, a2);
  lod_accumulate<64 >(GlobalGet{reinterpret_cast<const F3*>(cb2)}, p.x, p.y, p.z, a0, a1, a2);
  lod_accumulate<128>(GlobalGet{reinterpret_cast<const F3*>(cb3)}, p.x, p.y, p.z, a0, a1, a2);
  lod_accumulate<256>(GlobalGet{reinterpret_cast<const F3*>(cb4)}, p.x, p.y, p.z, a0, a1, a2);

  F3 o; o.x = a0; o.y = a1; o.z = a2;
  reinterpret_cast<F3*>(out)[n] = o;  // global_store_b96
}

}  // namespace

extern "C" void kernel_launch(void* const* d_in, const int* in_sizes, int n_in,
                              void* d_out, int out_size, void* d_ws, size_t ws_size,
                              hipStream_t stream) {
  (void)n_in; (void)d_ws; (void)ws_size; (void)out_size;
  const float* pts = (const float*)d_in[0];
  const float* cb0 = (const float*)d_in[1];
  const float* cb1 = (const float*)d_in[2];
  const float* cb2 = (const float*)d_in[3];
  const float* cb3 = (const float*)d_in[4];
  const float* cb4 = (const float*)d_in[5];
  const int N = in_sizes[0] / 3;  // pts is [N,3]
  const int blocks = (N + BLOCK - 1) / BLOCK;
  dense_grid_fused<<<blocks, BLOCK, 0, stream>>>(pts, cb0, cb1, cb2, cb3, cb4,
                                                 (float*)d_out, N);
}